// ChebConvGNN_11940009083288
// MI455X (gfx1250) — compile-verified
//
#include <hip/hip_runtime.h>

// ChebConv GNN forward for MI455X (gfx1250, wave32).
// Dense 256x256 GEMMs use V_WMMA_F32_16X16X4_F32 (fp32 matrix pipe, matches
// reference precision); sparse propagation uses L2-resident atomic scatter.
// GEMM: each wave computes a 16x64 output strip (4 WMMA accumulators) so one
// LDS A-fetch feeds 4 WMMAs; 4 waves/block share one A tile for 16 tiles.

#define FDIM 256
#define KORD 10
#define NMID 6
#define LDS_STRIDE 260   // 256 + 4 pad -> bank stride 4 on 64 banks

typedef __attribute__((ext_vector_type(2))) float v2f;
typedef __attribute__((ext_vector_type(8))) float v8f;

// ---------------- elementwise helpers ----------------

__global__ void k_fill(float* __restrict__ p, int n, float v) {
  int i = blockIdx.x * blockDim.x + threadIdx.x;
  if (i < n) p[i] = v;
}

__global__ void k_biasinit(float* __restrict__ acc, const float* __restrict__ bias, int n) {
  int i = blockIdx.x * blockDim.x + threadIdx.x;
  if (i < n) acc[i] = bias[i & (FDIM - 1)];
}

__global__ void k_outinit(float* __restrict__ out, const float* __restrict__ bias, int n) {
  int i = blockIdx.x * blockDim.x + threadIdx.x;
  if (i < n) out[i] = bias[0];
}

__global__ void k_relu(const float* __restrict__ a, float* __restrict__ h, int n) {
  int i = blockIdx.x * blockDim.x + threadIdx.x;
  if (i < n) h[i] = fmaxf(a[i], 0.0f);
}

// P = 2*P - T0   (Chebyshev recurrence tail)
__global__ void k_recur(float* __restrict__ P, const float* __restrict__ T0, int n) {
  int i = blockIdx.x * blockDim.x + threadIdx.x;
  if (i < n) P[i] = 2.0f * P[i] - T0[i];
}

// ---------------- graph normalization ----------------

__global__ void k_deg(const int* __restrict__ src, float* __restrict__ deg, int E) {
  int e = blockIdx.x * blockDim.x + threadIdx.x;
  if (e < E) atomicAdd(&deg[src[e]], 1.0f);
}

__global__ void k_norm(const int* __restrict__ src, const int* __restrict__ dst,
                       const float* __restrict__ deg, float* __restrict__ nrm, int E) {
  int e = blockIdx.x * blockDim.x + threadIdx.x;
  if (e >= E) return;
  float ds = deg[src[e]], dd = deg[dst[e]];
  float is = ds > 0.0f ? rsqrtf(fmaxf(ds, 1.0f)) : 0.0f;
  float id = dd > 0.0f ? rsqrtf(fmaxf(dd, 1.0f)) : 0.0f;
  nrm[e] = -is * id;
}

// ---------------- sparse propagation ----------------

// F=3 (first layer): one thread per edge
__global__ void k_prop3(const float* __restrict__ T, const int* __restrict__ src,
                        const int* __restrict__ dst, const float* __restrict__ nrm,
                        float* __restrict__ P, int E) {
  int e = blockIdx.x * blockDim.x + threadIdx.x;
  if (e >= E) return;
  int s = src[e], d = dst[e];
  float w = nrm[e];
  atomicAdd(&P[d * 3 + 0], w * T[s * 3 + 0]);
  atomicAdd(&P[d * 3 + 1], w * T[s * 3 + 1]);
  atomicAdd(&P[d * 3 + 2], w * T[s * 3 + 2]);
}

// F=256: one 256-thread block per edge; s/d/w are block-uniform (scalarized)
__global__ __launch_bounds__(256)
void k_prop256(const float* __restrict__ T, const int* __restrict__ src,
               const int* __restrict__ dst, const float* __restrict__ nrm,
               float* __restrict__ P) {
  int e = blockIdx.x;
  int f = threadIdx.x;
  int s = src[e], d = dst[e];
  float w = nrm[e];
  atomicAdd(&P[(size_t)d * FDIM + f], w * T[(size_t)s * FDIM + f]);
}

// ---------------- dense GEMMs ----------------

// First layer: acc[n,f] += sum_{c<3} T[n,c] * Wk[c,f]
__global__ void k_gemm3(const float* __restrict__ T, const float* __restrict__ Wk,
                        float* __restrict__ acc, int n) {
  int i = blockIdx.x * blockDim.x + threadIdx.x;
  if (i >= n) return;
  int node = i >> 8;
  int f = i & (FDIM - 1);
  const float* t = T + (size_t)node * 3;
  acc[i] += t[0] * Wk[f] + t[1] * Wk[FDIM + f] + t[2] * Wk[2 * FDIM + f];
}

// Mid layers: acc(Nx256) += T(Nx256) @ Wk(256x256), fp32 WMMA.
// Block = 4 waves, one node-tile (tm) per block; wave w computes cols
// [w*64, w*64+64) with 4 accumulators. A tile staged once in LDS.
__global__ __launch_bounds__(128)
void k_gemm_wmma(const float* __restrict__ T, const float* __restrict__ Wk,
                 float* __restrict__ acc) {
  __shared__ float lds[16 * LDS_STRIDE];
  const int wave = threadIdx.x >> 5;
  const int lane = threadIdx.x & 31;
  const int tm = blockIdx.x;

  // Stage A tile (16 rows x 256 cols of T) into LDS, float4 transfers.
  {
    int t = threadIdx.x;
    int r = t >> 3;            // 0..15
    int c = (t & 7) * 32;      // 8 threads per row, 32 floats each
    const float4* g = (const float4*)(T + (size_t)(tm * 16 + r) * FDIM + c);
    float4* l = (float4*)(lds + r * LDS_STRIDE + c);
#pragma unroll
    for (int j = 0; j < 8; ++j) l[j] = g[j];
  }
  __syncthreads();

  // A (16x4 fp32): lane<16 -> row=lane, K pair {0,1}; lane>=16 -> K pair {2,3}
  const int ra = lane & 15;
  const int kg = (lane >> 4) << 1;                // 0 or 2
  const float* arow = lds + ra * LDS_STRIDE + kg;
  // B (4x16 fp32): VGPR v, lane half -> row K = kg+v ; col = lane&15
  const int ncol0 = wave * 64 + (lane & 15);      // first of 4 column groups
  const float* bcol = Wk + ncol0;

  // C (16x16 fp32): VGPR v -> M = v + 8*(lane>=16); N = lane&15
  const int crow0 = tm * 16 + ((lane >> 4) << 3);

  v8f c0, c1, c2, c3;
#pragma unroll
  for (int v = 0; v < 8; ++v) {
    const float* ar = acc + (size_t)(crow0 + v) * FDIM + ncol0;
    c0[v] = ar[0];
    c1[v] = ar[16];
    c2[v] = ar[32];
    c3[v] = ar[48];
  }

#pragma unroll 4
  for (int kk = 0; kk < FDIM; kk += 4) {
    v2f a;
    a.x = arow[kk];
    a.y = arow[kk + 1];
    const float* b0 = bcol + (size_t)(kk + kg) * FDIM;
    const float* b1 = b0 + FDIM;
    v2f b;
    b.x = b0[0];
    b.y = b1[0];
    c0 = __builtin_amdgcn_wmma_f32_16x16x4_f32(false, a, false, b, (short)0, c0, false, false);
    b.x = b0[16];
    b.y = b1[16];
    c1 = __builtin_amdgcn_wmma_f32_16x16x4_f32(false, a, false, b, (short)0, c1, false, false);
    b.x = b0[32];
    b.y = b1[32];
    c2 = __builtin_amdgcn_wmma_f32_16x16x4_f32(false, a, false, b, (short)0, c2, false, false);
    b.x = b0[48];
    b.y = b1[48];
    c3 = __builtin_amdgcn_wmma_f32_16x16x4_f32(false, a, false, b, (short)0, c3, false, false);
  }

#pragma unroll
  for (int v = 0; v < 8; ++v) {
    float* ar = acc + (size_t)(crow0 + v) * FDIM + ncol0;
    ar[0]  = c0[v];
    ar[16] = c1[v];
    ar[32] = c2[v];
    ar[48] = c3[v];
  }
}

// Last layer (Fout=1): wave-per-node dot product with shuffle reduction
__global__ __launch_bounds__(256)
void k_dot(const float* __restrict__ T, const float* __restrict__ Wk,
           float* __restrict__ out, int n) {
  int gw = (blockIdx.x * blockDim.x + threadIdx.x) >> 5;
  int lane = threadIdx.x & 31;
  if (gw >= n) return;
  const float* t = T + (size_t)gw * FDIM + lane * 8;
  const float* w = Wk + lane * 8;
  float s = 0.0f;
#pragma unroll
  for (int j = 0; j < 8; ++j) s += t[j] * w[j];
#pragma unroll
  for (int off = 16; off; off >>= 1) s += __shfl_xor(s, off, 32);
  if (lane == 0) out[gw] += s;
}

// ---------------- host orchestration ----------------

extern "C" void kernel_launch(void* const* d_in, const int* in_sizes, int n_in,
                              void* d_out, int out_size, void* d_ws, size_t ws_size,
                              hipStream_t stream) {
  const float* x  = (const float*)d_in[0];
  const int*   ei = (const int*)d_in[1];
  const float* Wf = (const float*)d_in[2];
  const float* bf = (const float*)d_in[3];
  const float* Wm = (const float*)d_in[4];
  const float* bm = (const float*)d_in[5];
  const float* Wl = (const float*)d_in[6];
  const float* bl = (const float*)d_in[7];
  (void)n_in; (void)out_size; (void)ws_size;

  const int N = in_sizes[0] / 3;     // 50000 (multiple of 16)
  const int E = in_sizes[1] / 2;     // 800000
  const int* src = ei;
  const int* dst = ei + E;
  float* out = (float*)d_out;

  // workspace carve (floats): deg | norm | h | acc | B1 | B2   (~208 MB)
  float* ws = (float*)d_ws;
  size_t off = 0;
  auto carve = [&](size_t n) { float* p = ws + off; off += (n + 63) & ~(size_t)63; return p; };
  float* deg  = carve((size_t)N);
  float* nrm  = carve((size_t)E);
  float* h    = carve((size_t)N * FDIM);
  float* accm = carve((size_t)N * FDIM);
  float* B1   = carve((size_t)N * FDIM);
  float* B2   = carve((size_t)N * FDIM);

  const int TB = 256;
  auto cdiv = [](long a, long b) { return (int)((a + b - 1) / b); };
  const int nf = N * FDIM;
  const int gN = cdiv(nf, TB);
  const int gemm_blocks = N / 16;   // one node-tile per block, 4 waves cover N=256

  // graph normalization
  k_fill<<<cdiv(N, TB), TB, 0, stream>>>(deg, N, 0.0f);
  k_deg<<<cdiv(E, TB), TB, 0, stream>>>(src, deg, E);
  k_norm<<<cdiv(E, TB), TB, 0, stream>>>(src, dst, deg, nrm, E);

  float* rot[3] = {B1, B2, h};   // rotation targets; h is safe once layer input dead

  // ---- first layer (Fin=3) ----
  k_biasinit<<<gN, TB, 0, stream>>>(accm, bf, nf);
  {
    const float* T0 = x;
    const float* T1 = nullptr;
    for (int k = 0; k < KORD; ++k) {
      const float* Tk;
      if (k == 0) {
        Tk = x;
      } else {
        float* P = rot[(k - 1) % 3];          // only first 3*N floats used
        k_fill<<<cdiv(3L * N, TB), TB, 0, stream>>>(P, 3 * N, 0.0f);
        k_prop3<<<cdiv(E, TB), TB, 0, stream>>>(k == 1 ? x : T1, src, dst, nrm, P, E);
        if (k >= 2) { k_recur<<<cdiv(3L * N, TB), TB, 0, stream>>>(P, T0, 3 * N); T0 = T1; }
        T1 = P; Tk = P;
      }
      k_gemm3<<<gN, TB, 0, stream>>>(Tk, Wf + (size_t)k * 3 * FDIM, accm, nf);
    }
  }
  k_relu<<<gN, TB, 0, stream>>>(accm, h, nf);

  // ---- mid layers (Fin=Fout=256, WMMA) ----
  for (int i = 0; i < NMID; ++i) {
    k_biasinit<<<gN, TB, 0, stream>>>(accm, bm + (size_t)i * FDIM, nf);
    const float* T0 = h;
    const float* T1 = nullptr;
    for (int k = 0; k < KORD; ++k) {
      const float* Tk;
      if (k == 0) {
        Tk = h;
      } else {
        float* P = rot[(k - 1) % 3];          // k=3 overwrites h (input dead by then)
        k_fill<<<gN, TB, 0, stream>>>(P, nf, 0.0f);
        k_prop256<<<E, 256, 0, stream>>>(k == 1 ? h : T1, src, dst, nrm, P);
        if (k >= 2) { k_recur<<<gN, TB, 0, stream>>>(P, T0, nf); T0 = T1; }
        T1 = P; Tk = P;
      }
      k_gemm_wmma<<<gemm_blocks, 128, 0, stream>>>(
          Tk, Wm + ((size_t)i * KORD + k) * FDIM * FDIM, accm);
    }
    k_relu<<<gN, TB, 0, stream>>>(accm, h, nf);
  }

  // ---- last layer (Fout=1) ----
  k_outinit<<<cdiv(N, TB), TB, 0, stream>>>(out, bl, N);
  {
    const float* T0 = h;
    const float* T1 = nullptr;
    for (int k = 0; k < KORD; ++k) {
      const float* Tk;
      if (k == 0) {
        Tk = h;
      } else {
        float* P = rot[(k - 1) % 3];
        k_fill<<<gN, TB, 0, stream>>>(P, nf, 0.0f);
        k_prop256<<<E, 256, 0, stream>>>(k == 1 ? h : T1, src, dst, nrm, P);
        if (k >= 2) { k_recur<<<gN, TB, 0, stream>>>(P, T0, nf); T0 = T1; }
        T1 = P; Tk = P;
      }
      k_dot<<<cdiv((long)N * 32, 256), 256, 0, stream>>>(Tk, Wl + (size_t)k * FDIM, out, N);
    }
  }
}